// GATModel_26834955665917
// MI455X (gfx1250) — compile-verified
//
#include <hip/hip_runtime.h>
#include <math.h>

#define NNODES 100000
#define NEDGES 1600000
#define NB     512
#define F_IN   64
#define NH     4
#define DH     64
#define HD     256
#define EXTRA  16

typedef __attribute__((ext_vector_type(2))) float v2f;
typedef __attribute__((ext_vector_type(8))) float v8f;

__device__ __forceinline__ float selu_f(float x){
  const float scale = 1.0507009873554805f, alpha = 1.6732632423543772f;
  return scale * (x > 0.f ? x : alpha * (expf(x) - 1.f));
}
__device__ __forceinline__ float lrelu_f(float x){ return x > 0.f ? x : 0.2f * x; }

// Order-preserving float <-> uint mapping for atomicMax on floats
__device__ __forceinline__ unsigned f2mono(float f){
  unsigned u = __float_as_uint(f);
  return (u & 0x80000000u) ? ~u : (u | 0x80000000u);
}
__device__ __forceinline__ float mono2f(unsigned u){
  unsigned v = (u & 0x80000000u) ? (u & 0x7FFFFFFFu) : ~u;
  return __uint_as_float(v);
}

// ---------------- fp32 WMMA GEMM: Y[N,256] = X[N,K] @ W[K,256] ----------------
// One wave per 16x128 output strip. Per K-step: load A fragment once, load all
// 8 B fragments into distinct registers (clause-groupable), then 8 back-to-back
// V_WMMA_F32_16X16X4_F32 ops. A-reuse cuts X traffic 8x; split load/compute
// phases avoid per-WMMA s_wait_loadcnt serialization.
__global__ void gemm_wmma_f32(const float* __restrict__ X, const float* __restrict__ W,
                              float* __restrict__ Y, int K){
  int lane = threadIdx.x;            // 0..31
  int r0 = blockIdx.x * 16;          // row tile
  int c0 = blockIdx.y * 128;         // column strip (8 x 16-wide tiles)
  int half = lane >> 4;              // 0: K pair {0,1}, 1: K pair {2,3}
  int l16  = lane & 15;
  v8f acc[8] = {};
  const float* xrow = X + (size_t)(r0 + l16) * K;
  for (int k = 0; k < K; k += 4){
    int ka = k + 2 * half;
    v2f a;
    a.x = xrow[ka];
    a.y = xrow[ka + 1];
    const float* w0 = W + (size_t)ka * HD + c0 + l16;   // row ka
    const float* w1 = w0 + HD;                          // row ka+1
    v2f bv[8];
    #pragma unroll
    for (int c = 0; c < 8; ++c){
      bv[c].x = w0[c * 16];
      bv[c].y = w1[c * 16];
    }
    #pragma unroll
    for (int c = 0; c < 8; ++c){
      acc[c] = __builtin_amdgcn_wmma_f32_16x16x4_f32(false, a, false, bv[c], (short)0,
                                                     acc[c], false, false);
    }
  }
  // C/D layout: VGPR i -> row (i + 8*half), col = l16 (within each 16-col tile)
  #pragma unroll
  for (int c = 0; c < 8; ++c){
    #pragma unroll
    for (int i = 0; i < 8; ++i){
      Y[(size_t)(r0 + i + 8 * half) * HD + c0 + c * 16 + l16] = acc[c][i];
    }
  }
}

// ---------------- per-node attention logits: el/er [N,H] ----------------
__global__ void attn_logits_kernel(const float* __restrict__ h,
                                   const float* __restrict__ al,
                                   const float* __restrict__ ar,
                                   float* __restrict__ el, float* __restrict__ er){
  __shared__ float sl[HD];
  __shared__ float sr[HD];
  int n = blockIdx.x, t = threadIdx.x;          // t = head*64 + d
  float hv = h[(size_t)n * HD + t];
  sl[t] = hv * al[t];
  sr[t] = hv * ar[t];
  __syncthreads();
  for (int s = 32; s > 0; s >>= 1){
    if ((t & 63) < s){ sl[t] += sl[t + s]; sr[t] += sr[t + s]; }
    __syncthreads();
  }
  if ((t & 63) == 0){
    int hh = t >> 6;
    el[n * NH + hh] = sl[t];
    er[n * NH + hh] = sr[t];
  }
}

__global__ void fill_f32(float* __restrict__ p, float v, int n){
  int i = blockIdx.x * blockDim.x + threadIdx.x;
  if (i < n) p[i] = v;
}
__global__ void fill_u32(unsigned* __restrict__ p, unsigned v, int n){
  int i = blockIdx.x * blockDim.x + threadIdx.x;
  if (i < n) p[i] = v;
}

// ---------------- edge phase ----------------
__global__ void edge_max_kernel(const int* __restrict__ src, const int* __restrict__ dst,
                                const float* __restrict__ el, const float* __restrict__ er,
                                unsigned* __restrict__ mmax, int total){
  int i = blockIdx.x * blockDim.x + threadIdx.x;
  if (i >= total) return;
  int e = i >> 2, hh = i & 3;
  int s = src[e], d = dst[e];
  float v = lrelu_f(el[s * NH + hh] + er[d * NH + hh]);
  atomicMax(&mmax[d * NH + hh], f2mono(v));
}

__global__ void edge_exp_kernel(const int* __restrict__ src, const int* __restrict__ dst,
                                const float* __restrict__ el, const float* __restrict__ er,
                                const unsigned* __restrict__ mmax,
                                float* __restrict__ exb, float* __restrict__ den, int total){
  int i = blockIdx.x * blockDim.x + threadIdx.x;
  if (i >= total) return;
  int e = i >> 2, hh = i & 3;
  int s = src[e], d = dst[e];
  float v = lrelu_f(el[s * NH + hh] + er[d * NH + hh]);
  float m = mono2f(mmax[d * NH + hh]);
  if (!(m > -1e30f && m < 1e30f)) m = 0.f;      // mirror reference isfinite guard
  float ex = expf(v - m);
  exb[i] = ex;                                  // i == e*NH + hh
  atomicAdd(&den[d * NH + hh], ex);
}

__global__ void edge_scatter_kernel(const int* __restrict__ src, const int* __restrict__ dst,
                                    const float* __restrict__ hbuf,
                                    const float* __restrict__ exb,
                                    const float* __restrict__ den,
                                    float* __restrict__ obuf, int total){
  int i = blockIdx.x * blockDim.x + threadIdx.x;
  if (i >= total) return;
  int e = i >> 2, hh = i & 3;
  int s = src[e], d = dst[e];
  float alpha = exb[i] / fmaxf(den[d * NH + hh], 1e-9f);
  const float4* hv = (const float4*)(hbuf + (size_t)s * HD + hh * DH);
  float* ob = obuf + (size_t)d * HD + hh * DH;
  #pragma unroll
  for (int j = 0; j < 16; ++j){
    float4 v = hv[j];
    atomicAdd(ob + 4 * j + 0, v.x * alpha);
    atomicAdd(ob + 4 * j + 1, v.y * alpha);
    atomicAdd(ob + 4 * j + 2, v.z * alpha);
    atomicAdd(ob + 4 * j + 3, v.w * alpha);
  }
}

__global__ void finalize_kernel(const float* __restrict__ obuf, const float* __restrict__ bias,
                                float* __restrict__ xbuf, int n){
  int i = blockIdx.x * blockDim.x + threadIdx.x;
  if (i < n) xbuf[i] = selu_f(obuf[i] + bias[i & (HD - 1)]);
}

// ---------------- readout ----------------
__global__ void readout_kernel(const float* __restrict__ x, const float* __restrict__ sw,
                               const float* __restrict__ sb, const int* __restrict__ node_graph,
                               float* __restrict__ num, float* __restrict__ deng){
  __shared__ float sd[HD];
  int n = blockIdx.x, t = threadIdx.x;
  float xv = x[(size_t)n * HD + t];
  sd[t] = xv * sw[t];
  __syncthreads();
  for (int s = 128; s > 0; s >>= 1){
    if (t < s) sd[t] += sd[t + s];
    __syncthreads();
  }
  float w = 1.f / (1.f + expf(-(sd[0] + sb[0])));
  int g = node_graph[n];
  atomicAdd(&num[(size_t)g * HD + t], w * xv);
  if (t == 0) atomicAdd(&deng[g], w);
}

// ---------------- small MLP head ----------------
__global__ void mlp1_kernel(const float* __restrict__ num, const float* __restrict__ deng,
                            const float* __restrict__ fg, const float* __restrict__ lw1,
                            const float* __restrict__ lb1, float* __restrict__ y1){
  int i = blockIdx.x * blockDim.x + threadIdx.x;
  if (i >= NB * 128) return;
  int b = i >> 7, j = i & 127;
  float dn = fmaxf(deng[b], 1e-9f);
  float acc = lb1[j];
  const float* nb = num + (size_t)b * HD;
  for (int k = 0; k < HD; ++k) acc += (nb[k] / dn) * lw1[k * 128 + j];
  const float* fb = fg + b * EXTRA;
  for (int k = 0; k < EXTRA; ++k) acc += fb[k] * lw1[(HD + k) * 128 + j];
  y1[i] = selu_f(acc);
}

__global__ void mlp2_kernel(const float* __restrict__ y1, const float* __restrict__ lw2,
                            const float* __restrict__ lb2, float* __restrict__ y2){
  int i = blockIdx.x * blockDim.x + threadIdx.x;
  if (i >= NB * 64) return;
  int b = i >> 6, j = i & 63;
  float acc = lb2[j];
  const float* yb = y1 + b * 128;
  for (int k = 0; k < 128; ++k) acc += yb[k] * lw2[k * 64 + j];
  y2[i] = selu_f(acc);
}

__global__ void mlp3_kernel(const float* __restrict__ y2, const float* __restrict__ lw3,
                            const float* __restrict__ lb3, float* __restrict__ out){
  int b = blockIdx.x * blockDim.x + threadIdx.x;
  if (b >= NB) return;
  float acc = lb3[0];
  const float* yb = y2 + b * 64;
  for (int k = 0; k < 64; ++k) acc += yb[k] * lw3[k];
  out[b] = acc;
}

static inline int nblk(int n){ return (n + 255) / 256; }

extern "C" void kernel_launch(void* const* d_in, const int* in_sizes, int n_in,
                              void* d_out, int out_size, void* d_ws, size_t ws_size,
                              hipStream_t stream) {
  const int*   src        = (const int*)  d_in[0];
  const int*   dst        = (const int*)  d_in[1];
  const int*   node_graph = (const int*)  d_in[2];
  const float* feats_node = (const float*)d_in[3];
  const float* feats_graph= (const float*)d_in[4];
  const float* W [3] = {(const float*)d_in[5],  (const float*)d_in[9],  (const float*)d_in[13]};
  const float* AL[3] = {(const float*)d_in[6],  (const float*)d_in[10], (const float*)d_in[14]};
  const float* AR[3] = {(const float*)d_in[7],  (const float*)d_in[11], (const float*)d_in[15]};
  const float* BG[3] = {(const float*)d_in[8],  (const float*)d_in[12], (const float*)d_in[16]};
  const float* sw  = (const float*)d_in[17];
  const float* sb  = (const float*)d_in[18];
  const float* lw1 = (const float*)d_in[19];
  const float* lb1 = (const float*)d_in[20];
  const float* lw2 = (const float*)d_in[21];
  const float* lb2 = (const float*)d_in[22];
  const float* lw3 = (const float*)d_in[23];
  const float* lb3 = (const float*)d_in[24];

  float* ws = (float*)d_ws;
  size_t off = 0;
  float*    xbuf  = ws + off; off += (size_t)NNODES * HD;   // 25.6M
  float*    hbuf  = ws + off; off += (size_t)NNODES * HD;   // 25.6M
  float*    obuf  = ws + off; off += (size_t)NNODES * HD;   // 25.6M
  float*    elb   = ws + off; off += (size_t)NNODES * NH;
  float*    erb   = ws + off; off += (size_t)NNODES * NH;
  unsigned* mmaxb = (unsigned*)(ws + off); off += (size_t)NNODES * NH;
  float*    denb  = ws + off; off += (size_t)NNODES * NH;
  float*    exb   = ws + off; off += (size_t)NEDGES * NH;   // 6.4M
  float*    num   = ws + off; off += (size_t)NB * HD;
  float*    deng  = ws + off; off += (size_t)NB;
  float*    y1    = ws + off; off += (size_t)NB * 128;
  float*    y2    = ws + off; off += (size_t)NB * 64;

  const int totEH = NEDGES * NH;        // 6.4M
  const int totNH = NNODES * NH;        // 400k
  const int totND = NNODES * HD;        // 25.6M

  const float* xin = feats_node;
  int K = F_IN;
  for (int layer = 0; layer < 3; ++layer){
    dim3 ggrid(NNODES / 16, HD / 128);  // 6250 x 2 strips of 16x128
    gemm_wmma_f32<<<ggrid, 32, 0, stream>>>(xin, W[layer], hbuf, K);
    attn_logits_kernel<<<NNODES, 256, 0, stream>>>(hbuf, AL[layer], AR[layer], elb, erb);
    fill_u32<<<nblk(totNH), 256, 0, stream>>>(mmaxb, 0x007FFFFFu, totNH);  // f2mono(-inf)
    fill_f32<<<nblk(totNH), 256, 0, stream>>>(denb, 0.f, totNH);
    fill_f32<<<nblk(totND), 256, 0, stream>>>(obuf, 0.f, totND);
    edge_max_kernel    <<<nblk(totEH), 256, 0, stream>>>(src, dst, elb, erb, mmaxb, totEH);
    edge_exp_kernel    <<<nblk(totEH), 256, 0, stream>>>(src, dst, elb, erb, mmaxb, exb, denb, totEH);
    edge_scatter_kernel<<<nblk(totEH), 256, 0, stream>>>(src, dst, hbuf, exb, denb, obuf, totEH);
    finalize_kernel    <<<nblk(totND), 256, 0, stream>>>(obuf, BG[layer], xbuf, totND);
    xin = xbuf; K = HD;
  }

  fill_f32<<<nblk(NB * HD), 256, 0, stream>>>(num, 0.f, NB * HD);
  fill_f32<<<nblk(NB), 256, 0, stream>>>(deng, 0.f, NB);
  readout_kernel<<<NNODES, 256, 0, stream>>>(xbuf, sw, sb, node_graph, num, deng);
  mlp1_kernel<<<nblk(NB * 128), 256, 0, stream>>>(num, deng, feats_graph, lw1, lb1, y1);
  mlp2_kernel<<<nblk(NB * 64), 256, 0, stream>>>(y1, lw2, lb2, y2);
  mlp3_kernel<<<nblk(NB), 256, 0, stream>>>(y2, lw3, lb3, (float*)d_out);
}